// captioner_4793183502898
// MI455X (gfx1250) — compile-verified
//
#include <hip/hip_runtime.h>
#include <math.h>

typedef float v2f __attribute__((ext_vector_type(2)));
typedef float v4f __attribute__((ext_vector_type(4)));
typedef float v8f __attribute__((ext_vector_type(8)));

#define BN_EPS 1e-5f
#define NTHREADS 128

__global__ __launch_bounds__(NTHREADS)
void captioner_fused_kernel(
    const float* __restrict__ x,        // (3,26,26)
    const float* __restrict__ hidden0,  // (25)
    const float* __restrict__ c1_w, const float* __restrict__ c1_b,
    const float* __restrict__ bn1_g, const float* __restrict__ bn1_b,
    const float* __restrict__ bn1_m, const float* __restrict__ bn1_v,
    const float* __restrict__ c2_w, const float* __restrict__ c2_b,
    const float* __restrict__ bn2_g, const float* __restrict__ bn2_b,
    const float* __restrict__ bn2_m, const float* __restrict__ bn2_v,
    const float* __restrict__ wih, const float* __restrict__ whh,
    const float* __restrict__ bih, const float* __restrict__ bhh,
    const float* __restrict__ out_w, const float* __restrict__ out_b,
    float* __restrict__ out)            // (25,100)
{
    __shared__ float xl[3 * 26 * 26];   // 2028 input image
    __shared__ float p1[4 * 12 * 12];   // 576  pooled conv1
    __shared__ float xseq[32 * 16];     // [pos(pad 32)][feat 16], zero-padded
    __shared__ float Xp[32 * 32];       // [t][j] = Wih@xt + bih + bhh (pads=0)
    __shared__ float H[32 * 32];        // [t][j] hidden states (padded)
    __shared__ float hbuf[32];          // current hidden (pads stay 0)
    __shared__ float wl[32 * 16];       // wih zero-padded to 32 rows
    __shared__ float owl[112 * 28];     // out_w zero-padded to 112x28
    __shared__ float whl[32 * 28];      // whh zero-padded to 32x28
    __shared__ float bsum[32];          // bih + bhh, zero-padded
    __shared__ float obl[112];          // out_b staged
    __shared__ float Ol[32 * 112];      // staged GEMM-2 result [t][o], padded
    __shared__ float c1wl[108], c2wl[576];
    __shared__ float s1l[4], t1l[4], s2l[16], t2l[16];

    const int tid  = threadIdx.x;
    const int lane = tid & 31;
    const int wid  = tid >> 5;          // 4 waves

    // ---- init: stage inputs into LDS with zero padding (one write/loc) ----
    for (int i = tid; i < 3 * 26 * 26; i += NTHREADS) xl[i] = x[i];
    for (int i = tid; i < 32 * 16; i += NTHREADS) {
        xseq[i] = 0.f;
        int m = i >> 4;
        wl[i] = (m < 25) ? wih[i] : 0.f;
    }
    for (int i = tid; i < 32 * 32; i += NTHREADS) H[i] = 0.f;
    for (int i = tid; i < 112 * 28; i += NTHREADS) {
        int m = i / 28, k = i - m * 28;
        owl[i] = (m < 100 && k < 25) ? out_w[m * 25 + k] : 0.f;
    }
    for (int i = tid; i < 32 * 28; i += NTHREADS) {
        int m = i / 28, k = i - m * 28;
        whl[i] = (m < 25 && k < 25) ? whh[m * 25 + k] : 0.f;
    }
    for (int i = tid; i < 108; i += NTHREADS) c1wl[i] = c1_w[i];
    for (int i = tid; i < 576; i += NTHREADS) c2wl[i] = c2_w[i];
    if (tid < 32) {
        hbuf[tid] = (tid < 25) ? hidden0[tid] : 0.f;
        bsum[tid] = (tid < 25) ? (bih[tid] + bhh[tid]) : 0.f;
    }
    if (tid >= 32 && tid < 32 + 4) {     // fold BN1 + conv bias
        int c = tid - 32;
        float s = bn1_g[c] * rsqrtf(bn1_v[c] + BN_EPS);
        s1l[c] = s;
        t1l[c] = c1_b[c] * s + bn1_b[c] - bn1_m[c] * s;
    }
    if (tid >= 64 && tid < 64 + 16) {    // fold BN2 + conv bias
        int c = tid - 64;
        float s = bn2_g[c] * rsqrtf(bn2_v[c] + BN_EPS);
        s2l[c] = s;
        t2l[c] = c2_b[c] * s + bn2_b[c] - bn2_m[c] * s;
    }
    if (tid >= 96 && tid < 96 + 16) {    // out_b staged
        for (int o = tid - 96; o < 100; o += 16) obl[o] = out_b[o];
    }
    __syncthreads();

    // ---- conv1 (3->4, 3x3) + BN + ReLU + maxpool2 -> p1 (4,12,12) ----
    for (int idx = tid; idx < 4 * 12 * 12; idx += NTHREADS) {
        int c1  = idx / 144;
        int rem = idx - c1 * 144;
        int pi  = rem / 12, pj = rem % 12;
        float s = s1l[c1], sh = t1l[c1];
        float best = -INFINITY;
        for (int dr2 = 0; dr2 < 2; ++dr2)
            for (int dc2 = 0; dc2 < 2; ++dc2) {
                int R = 2 * pi + dr2, C = 2 * pj + dc2;
                float acc = 0.f;
                #pragma unroll
                for (int ic = 0; ic < 3; ++ic)
                    #pragma unroll
                    for (int dr = 0; dr < 3; ++dr)
                        #pragma unroll
                        for (int dc = 0; dc < 3; ++dc)
                            acc += xl[ic * 676 + (R + dr) * 26 + (C + dc)] *
                                   c1wl[((c1 * 3 + ic) * 3 + dr) * 3 + dc];
                best = fmaxf(best, fmaxf(acc * s + sh, 0.f));
            }
        p1[idx] = best;
    }
    __syncthreads();

    // ---- conv2 (4->16, 3x3) + BN + ReLU + maxpool2 -> xseq[pos][c2] ----
    for (int idx = tid; idx < 16 * 25; idx += NTHREADS) {
        int c2  = idx / 25;
        int pos = idx - c2 * 25;
        int r = pos / 5, c = pos % 5;
        float s = s2l[c2], sh = t2l[c2];
        float best = -INFINITY;
        for (int dr2 = 0; dr2 < 2; ++dr2)
            for (int dc2 = 0; dc2 < 2; ++dc2) {
                int R = 2 * r + dr2, C = 2 * c + dc2;
                float acc = 0.f;
                #pragma unroll
                for (int ic = 0; ic < 4; ++ic)
                    #pragma unroll
                    for (int dr = 0; dr < 3; ++dr)
                        #pragma unroll
                        for (int dc = 0; dc < 3; ++dc)
                            acc += p1[ic * 144 + (R + dr) * 12 + (C + dc)] *
                                   c2wl[((c2 * 4 + ic) * 3 + dr) * 3 + dc];
                best = fmaxf(best, fmaxf(acc * s + sh, 0.f));
            }
        xseq[pos * 16 + c2] = best;
    }
    __syncthreads();

    // ---- WMMA GEMM 1: Xp[t][j] = (Wih @ xseq^T)[j][t] + bsum[j] ----
    // Padded M=32(j), N=32(t), K=16. One tile per wave; branch-free epilogue.
    // All pads are exact zeros, so Xp pads come out exactly 0.
    {
        const int M0 = (wid & 1) * 16, N0 = (wid >> 1) * 16;
        const int half = lane >> 4, lm = lane & 15;
        const int m = M0 + lm;                    // Wih row (j), padded
        const int n = N0 + lm;                    // step t, padded
        v8f c = {};
        #pragma unroll
        for (int k0 = 0; k0 < 16; k0 += 4) {
            int ka = k0 + 2 * half;               // even -> 8B aligned
            v2f a = *(const v2f*)&wl[m * 16 + ka];
            v2f b = *(const v2f*)&xseq[n * 16 + ka];
            c = __builtin_amdgcn_wmma_f32_16x16x4_f32(
                    false, a, false, b, (short)0, c, false, false);
        }
        const int base = M0 + half * 8;           // consecutive j for r=0..7
        v4f lo, hi;
        #pragma unroll
        for (int r = 0; r < 4; ++r) {
            lo[r] = c[r]     + bsum[base + r];
            hi[r] = c[4 + r] + bsum[base + 4 + r];
        }
        *(v4f*)&Xp[n * 32 + base]     = lo;       // 16B-aligned b128 stores
        *(v4f*)&Xp[n * 32 + base + 4] = hi;
    }
    __syncthreads();

    // ---- sequential RNN recurrence, wave 0 only. Whh row hoisted into
    // registers once; hbuf read as 7 broadcast b128 loads per step.
    // Lanes 25..31 compute tanh(0)=0 (all pads exact zeros) -> harmless.
    if (wid == 0) {
        float wreg[28];
        #pragma unroll
        for (int k = 0; k < 28; ++k) wreg[k] = whl[lane * 28 + k];
        for (int t = 0; t < 25; ++t) {
            float acc = Xp[t * 32 + lane];
            #pragma unroll
            for (int k0 = 0; k0 < 28; k0 += 4) {
                v4f hv = *(const v4f*)&hbuf[k0];  // broadcast, conflict-free
                acc += wreg[k0]     * hv[0];
                acc += wreg[k0 + 1] * hv[1];
                acc += wreg[k0 + 2] * hv[2];
                acc += wreg[k0 + 3] * hv[3];
            }
            float hn = tanhf(acc);
            __builtin_amdgcn_wave_barrier();      // order reads before write
            hbuf[lane] = hn;
            H[t * 32 + lane] = hn;
            __builtin_amdgcn_wave_barrier();      // order write before next read
        }
    }
    __syncthreads();

    // ---- WMMA GEMM 2: Ol[t][o] = (out_w @ H^T)[o][t]  (bias+tanh later) ----
    // Padded M=112(o, 7 tiles), N=32(t, 2 tiles), K=28. 14 tiles / 4 waves,
    // branch-free b128 epilogue into padded staging buffer.
    {
        const int half = lane >> 4, lm = lane & 15;
        for (int tile = wid; tile < 14; tile += 4) {
            int M0 = (tile % 7) * 16, N0 = (tile / 7) * 16;
            const int m = M0 + lm;                // output row o, padded
            const int n = N0 + lm;                // step t, padded
            v8f c = {};
            #pragma unroll
            for (int k0 = 0; k0 < 28; k0 += 4) {
                int ka = k0 + 2 * half;           // even -> 8B aligned
                v2f a = *(const v2f*)&owl[m * 28 + ka];
                v2f b = *(const v2f*)&H[n * 32 + ka];
                c = __builtin_amdgcn_wmma_f32_16x16x4_f32(
                        false, a, false, b, (short)0, c, false, false);
            }
            const int base = M0 + half * 8;       // consecutive o for r=0..7
            v4f lo, hi;
            #pragma unroll
            for (int r = 0; r < 4; ++r) { lo[r] = c[r]; hi[r] = c[4 + r]; }
            *(v4f*)&Ol[n * 112 + base]     = lo;  // 16B-aligned b128 stores
            *(v4f*)&Ol[n * 112 + base + 4] = hi;
        }
    }
    __syncthreads();

    // ---- epilogue: bias + tanh + coalesced store of the valid 25x100 ----
    for (int i = tid; i < 25 * 100; i += NTHREADS) {
        int t = i / 100, o = i - t * 100;
        out[i] = tanhf(Ol[t * 112 + o] + obl[o]);
    }
}

extern "C" void kernel_launch(void* const* d_in, const int* in_sizes, int n_in,
                              void* d_out, int out_size, void* d_ws, size_t ws_size,
                              hipStream_t stream) {
    (void)in_sizes; (void)n_in; (void)out_size; (void)d_ws; (void)ws_size;
    const float* x       = (const float*)d_in[0];
    const float* hidden0 = (const float*)d_in[1];
    const float* c1_w    = (const float*)d_in[2];
    const float* c1_b    = (const float*)d_in[3];
    const float* bn1_g   = (const float*)d_in[4];
    const float* bn1_b   = (const float*)d_in[5];
    const float* bn1_m   = (const float*)d_in[6];
    const float* bn1_v   = (const float*)d_in[7];
    const float* c2_w    = (const float*)d_in[8];
    const float* c2_b    = (const float*)d_in[9];
    const float* bn2_g   = (const float*)d_in[10];
    const float* bn2_b   = (const float*)d_in[11];
    const float* bn2_m   = (const float*)d_in[12];
    const float* bn2_v   = (const float*)d_in[13];
    const float* wih     = (const float*)d_in[14];
    const float* whh     = (const float*)d_in[15];
    const float* bih     = (const float*)d_in[16];
    const float* bhh     = (const float*)d_in[17];
    const float* out_w   = (const float*)d_in[18];
    const float* out_b   = (const float*)d_in[19];
    float* out = (float*)d_out;

    captioner_fused_kernel<<<1, NTHREADS, 0, stream>>>(
        x, hidden0, c1_w, c1_b, bn1_g, bn1_b, bn1_m, bn1_v,
        c2_w, c2_b, bn2_g, bn2_b, bn2_m, bn2_v,
        wih, whh, bih, bhh, out_w, out_b, out);
}